// TGNMemory_52922587021368
// MI455X (gfx1250) — compile-verified
//
#include <hip/hip_runtime.h>
#include <math.h>

#define N_NODES 200000
#define E_EDGES 100000
#define RAW 172
#define MEMW 100
#define TDIM 100
#define HSZ 100
#define MSGW 472  // 2*MEM + RAW + TDIM
#define MSG2 372  // message cols 100..471 (other-mem, raw, time enc)

// LDS strides (floats): even (8B-aligned b64 ds reads), stride%64 has gcd 2
// with 64 so 16 consecutive rows land on distinct bank pairs.
#define MEMS 102
#define MS2  374

typedef __attribute__((ext_vector_type(2))) float v2f;
typedef __attribute__((ext_vector_type(8))) float v8f;

__global__ void init_winner(unsigned long long* win) {
  int i = blockIdx.x * blockDim.x + threadIdx.x;
  if (i < N_NODES) win[i] = 0ull;
}

// Packed key ((t+1)<<32)|eid: max key == max t, tie-break max eid ==
// reference segment_max(t) then segment_max(eid where t==max).
__global__ void winner_kernel(const int* __restrict__ src, const int* __restrict__ dst,
                              const int* __restrict__ t, unsigned long long* win) {
  int e = blockIdx.x * blockDim.x + threadIdx.x;
  if (e >= E_EDGES) return;
  unsigned long long tv = (unsigned long long)(unsigned int)(t[e] + 1);
  atomicMax(&win[src[e]], (tv << 32) | (unsigned long long)(unsigned int)e);
  atomicMax(&win[dst[e]], (tv << 32) | (unsigned long long)(unsigned int)(e + E_EDGES));
}

__device__ __forceinline__ float sigmoidf_(float x) { return 1.0f / (1.0f + expf(-x)); }

// 32-node tile per workgroup (128 threads = 4 waves sharing the LDS tile).
// Each wave owns column-tiles ct = waveid, waveid+4 of the 7 (16-wide) gate
// column tiles. Register blocking: 2 M-fragments (rows 0-15, 16-31) so every
// B fragment feeds two v_wmma_f32_16x16x4_f32.
__global__ void __launch_bounds__(128)
fused_gru(const float* __restrict__ memory, const float* __restrict__ raw_msg,
          const float* __restrict__ time_w, const float* __restrict__ time_b,
          const float* __restrict__ w_ih, const float* __restrict__ w_hh,
          const float* __restrict__ b_ih, const float* __restrict__ b_hh,
          const int* __restrict__ last_update, const int* __restrict__ src,
          const int* __restrict__ dst, const int* __restrict__ t,
          const unsigned long long* __restrict__ win,
          float* __restrict__ out_mem, float* __restrict__ out_lu) {
  __shared__ float lds_mem[32 * MEMS];   // memory[node] rows (unmasked)
  __shared__ float lds_msg2[32 * MS2];   // message cols 100..471 (0 if !has)
  __shared__ int   s_e[32];
  __shared__ int   s_other[32];
  __shared__ float s_hasf[32];
  __shared__ float s_dt[32];

  const int tid = threadIdx.x;
  const int node0 = blockIdx.x * 32;

  if (tid < 32) {
    int node = node0 + tid;
    unsigned long long key = win[node];
    int has = (key != 0ull) ? 1 : 0;
    int eid = (int)(key & 0xffffffffull);
    int e = (eid < E_EDGES) ? eid : eid - E_EDGES;
    int maxt = (int)(key >> 32) - 1;
    int other = node;
    if (has) other = (eid < E_EDGES) ? dst[e] : src[e]; else e = 0;
    s_e[tid] = e;
    s_other[tid] = other;
    s_hasf[tid] = (float)has;
    s_dt[tid] = has ? ((float)t[e] - (float)last_update[node]) : 0.0f;
    out_lu[node] = (float)(has ? maxt : last_update[node]);
  }
  __syncthreads();

  // 32 x 100 self-memory tile
  for (int idx = tid; idx < 32 * MEMW; idx += 128) {
    int i = idx / MEMW, c = idx - i * MEMW;
    lds_mem[i * MEMS + c] = memory[(size_t)(node0 + i) * MEMW + c];
  }
  // 32 x 372 message tail: [mem[other], raw_msg[e], cos(dt*w + b)], 0 if !has
  for (int idx = tid; idx < 32 * MSG2; idx += 128) {
    int i = idx / MSG2, c = idx - i * MSG2;
    float v = 0.0f;
    if (s_hasf[i] != 0.0f) {
      if (c < MEMW)            v = memory[(size_t)s_other[i] * MEMW + c];
      else if (c < MEMW + RAW) v = raw_msg[(size_t)s_e[i] * RAW + (c - MEMW)];
      else { int k = c - MEMW - RAW; v = cosf(s_dt[i] * time_w[k] + time_b[k]); }
    }
    lds_msg2[i * MS2 + c] = v;
  }
  __syncthreads();

  // WMMA 16x16x4 f32 lane mapping:
  //   A: lane L -> row M=L&15; VGPRs hold K = k0+{0,1} (L<16) / {2,3} (L>=16)
  //   B: lane L -> col N=L&15; same K split
  //   C/D: VGPR v -> row (v | (L>=16 ? 8:0)), col L&15
  const int lane = tid & 31;
  const int wave = tid >> 5;
  const int nlow = lane & 15;
  const int koff = (lane < 16) ? 0 : 2;
  const int hi8  = (lane < 16) ? 0 : 8;
  const float hasf_lo = s_hasf[nlow];        // mask for rows 0-15 fragment
  const float hasf_hi = s_hasf[nlow + 16];   // mask for rows 16-31 fragment

  for (int ct = wave; ct < 7; ct += 4) {
    const int c0 = ct * 16;
    const int col = c0 + nlow;
    const bool colok = (col < HSZ);
    const int cc = colok ? col : (HSZ - 1);  // clamp for safe addressing

    v8f gir0 = {}, giz0 = {}, gin0 = {}, ghr0 = {}, ghz0 = {}, ghn0 = {};
    v8f gir1 = {}, giz1 = {}, gin1 = {}, ghr1 = {}, ghz1 = {}, ghn1 = {};

    const float* pir = w_ih + (size_t)cc * MSGW;
    const float* piz = pir + (size_t)HSZ * MSGW;
    const float* pin = piz + (size_t)HSZ * MSGW;
    const float* phr = w_hh + (size_t)cc * HSZ;
    const float* phz = phr + (size_t)HSZ * HSZ;
    const float* phn = phz + (size_t)HSZ * HSZ;

    // Fused K in [0,100): gh (unmasked A) + gi self-memory block (A * has)
    for (int k0 = 0; k0 < HSZ; k0 += 4) {
      int k = k0 + koff;
      v2f alo; alo.x = lds_mem[nlow * MEMS + k];        alo.y = lds_mem[nlow * MEMS + k + 1];
      v2f ahi; ahi.x = lds_mem[(nlow + 16) * MEMS + k]; ahi.y = lds_mem[(nlow + 16) * MEMS + k + 1];
      v2f bir; bir.x = pir[k]; bir.y = pir[k + 1];
      v2f biz; biz.x = piz[k]; biz.y = piz[k + 1];
      v2f bin; bin.x = pin[k]; bin.y = pin[k + 1];
      v2f bhr; bhr.x = phr[k]; bhr.y = phr[k + 1];
      v2f bhz; bhz.x = phz[k]; bhz.y = phz[k + 1];
      v2f bhn; bhn.x = phn[k]; bhn.y = phn[k + 1];
      v2f mlo = alo * hasf_lo;
      v2f mhi = ahi * hasf_hi;
      ghr0 = __builtin_amdgcn_wmma_f32_16x16x4_f32(false, alo, false, bhr, (short)0, ghr0, false, false);
      ghr1 = __builtin_amdgcn_wmma_f32_16x16x4_f32(false, ahi, false, bhr, (short)0, ghr1, false, false);
      ghz0 = __builtin_amdgcn_wmma_f32_16x16x4_f32(false, alo, false, bhz, (short)0, ghz0, false, false);
      ghz1 = __builtin_amdgcn_wmma_f32_16x16x4_f32(false, ahi, false, bhz, (short)0, ghz1, false, false);
      ghn0 = __builtin_amdgcn_wmma_f32_16x16x4_f32(false, alo, false, bhn, (short)0, ghn0, false, false);
      ghn1 = __builtin_amdgcn_wmma_f32_16x16x4_f32(false, ahi, false, bhn, (short)0, ghn1, false, false);
      gir0 = __builtin_amdgcn_wmma_f32_16x16x4_f32(false, mlo, false, bir, (short)0, gir0, false, false);
      gir1 = __builtin_amdgcn_wmma_f32_16x16x4_f32(false, mhi, false, bir, (short)0, gir1, false, false);
      giz0 = __builtin_amdgcn_wmma_f32_16x16x4_f32(false, mlo, false, biz, (short)0, giz0, false, false);
      giz1 = __builtin_amdgcn_wmma_f32_16x16x4_f32(false, mhi, false, biz, (short)0, giz1, false, false);
      gin0 = __builtin_amdgcn_wmma_f32_16x16x4_f32(false, mlo, false, bin, (short)0, gin0, false, false);
      gin1 = __builtin_amdgcn_wmma_f32_16x16x4_f32(false, mhi, false, bin, (short)0, gin1, false, false);
    }

    // gi tail: K in [100,472)  (A = lds_msg2, already zero when !has)
    const float* qir = pir + MEMW;
    const float* qiz = piz + MEMW;
    const float* qin = pin + MEMW;
    for (int k0 = 0; k0 < MSG2; k0 += 4) {
      int k = k0 + koff;
      v2f alo; alo.x = lds_msg2[nlow * MS2 + k];        alo.y = lds_msg2[nlow * MS2 + k + 1];
      v2f ahi; ahi.x = lds_msg2[(nlow + 16) * MS2 + k]; ahi.y = lds_msg2[(nlow + 16) * MS2 + k + 1];
      v2f bir; bir.x = qir[k]; bir.y = qir[k + 1];
      v2f biz; biz.x = qiz[k]; biz.y = qiz[k + 1];
      v2f bin; bin.x = qin[k]; bin.y = qin[k + 1];
      gir0 = __builtin_amdgcn_wmma_f32_16x16x4_f32(false, alo, false, bir, (short)0, gir0, false, false);
      gir1 = __builtin_amdgcn_wmma_f32_16x16x4_f32(false, ahi, false, bir, (short)0, gir1, false, false);
      giz0 = __builtin_amdgcn_wmma_f32_16x16x4_f32(false, alo, false, biz, (short)0, giz0, false, false);
      giz1 = __builtin_amdgcn_wmma_f32_16x16x4_f32(false, ahi, false, biz, (short)0, giz1, false, false);
      gin0 = __builtin_amdgcn_wmma_f32_16x16x4_f32(false, alo, false, bin, (short)0, gin0, false, false);
      gin1 = __builtin_amdgcn_wmma_f32_16x16x4_f32(false, ahi, false, bin, (short)0, gin1, false, false);
    }

    const float bbir = b_ih[cc], bbiz = b_ih[cc + HSZ], bbin = b_ih[cc + 2 * HSZ];
    const float bbhr = b_hh[cc], bbhz = b_hh[cc + HSZ], bbhn = b_hh[cc + 2 * HSZ];

#pragma unroll
    for (int v = 0; v < 8; ++v) {
      // fragment 0: rows 0-15
      {
        int row = v + hi8;
        float r  = sigmoidf_((gir0[v] + bbir) + (ghr0[v] + bbhr));
        float z  = sigmoidf_((giz0[v] + bbiz) + (ghz0[v] + bbhz));
        float nn = tanhf((gin0[v] + bbin) + r * (ghn0[v] + bbhn));
        float h  = lds_mem[row * MEMS + cc];
        if (colok) out_mem[(size_t)(node0 + row) * HSZ + col] = (1.0f - z) * nn + z * h;
      }
      // fragment 1: rows 16-31
      {
        int row = 16 + v + hi8;
        float r  = sigmoidf_((gir1[v] + bbir) + (ghr1[v] + bbhr));
        float z  = sigmoidf_((giz1[v] + bbiz) + (ghz1[v] + bbhz));
        float nn = tanhf((gin1[v] + bbin) + r * (ghn1[v] + bbhn));
        float h  = lds_mem[row * MEMS + cc];
        if (colok) out_mem[(size_t)(node0 + row) * HSZ + col] = (1.0f - z) * nn + z * h;
      }
    }
  }
}

extern "C" void kernel_launch(void* const* d_in, const int* in_sizes, int n_in,
                              void* d_out, int out_size, void* d_ws, size_t ws_size,
                              hipStream_t stream) {
  const float* memory      = (const float*)d_in[0];
  const float* raw_msg     = (const float*)d_in[1];
  const float* time_w      = (const float*)d_in[2];
  const float* time_b      = (const float*)d_in[3];
  const float* w_ih        = (const float*)d_in[4];
  const float* w_hh        = (const float*)d_in[5];
  const float* b_ih        = (const float*)d_in[6];
  const float* b_hh        = (const float*)d_in[7];
  const int*   last_update = (const int*)d_in[8];
  const int*   src         = (const int*)d_in[9];
  const int*   dst         = (const int*)d_in[10];
  const int*   t           = (const int*)d_in[11];

  float* out_mem = (float*)d_out;
  float* out_lu  = out_mem + (size_t)N_NODES * MEMW;
  unsigned long long* win = (unsigned long long*)d_ws;  // N_NODES * 8 bytes

  init_winner<<<(N_NODES + 255) / 256, 256, 0, stream>>>(win);
  winner_kernel<<<(E_EDGES + 255) / 256, 256, 0, stream>>>(src, dst, t, win);
  fused_gru<<<N_NODES / 32, 128, 0, stream>>>(memory, raw_msg, time_w, time_b,
                                              w_ih, w_hh, b_ih, b_hh,
                                              last_update, src, dst, t, win,
                                              out_mem, out_lu);
}